// SoftRankNDCGLoss_2130303779112
// MI455X (gfx1250) — compile-verified
//
#include <hip/hip_runtime.h>
#include <hip/hip_bf16.h>

#define BATCH 128
#define LEN   1024
#define NT    256

typedef unsigned int v4u __attribute__((ext_vector_type(4)));
typedef int          v8i __attribute__((ext_vector_type(8)));
typedef int          v4i __attribute__((ext_vector_type(4)));

// LDS layout (all dynamic so the preds buffer is at LDS offset 0 for the TDM):
//   [0,      4096)  : preds row, 1024 f32
//   [4096,   6144)  : reduction buffer, 512 f32
//   [6144,   6176)  : histogram counts, 8 u32
#define SMEM_BYTES 6176

// Abramowitz-Stegun 7.1.26 erf, |err| <= 1.5e-7. Uses v_rcp_f32 + v_exp_f32.
__device__ __forceinline__ float erf_fast(float x) {
    float ax = __builtin_fabsf(x);
    float t  = __builtin_amdgcn_rcpf(__builtin_fmaf(0.3275911f, ax, 1.0f));
    float p  = __builtin_fmaf(t, 1.061405429f, -1.453152027f);
    p = __builtin_fmaf(t, p, 1.421413741f);
    p = __builtin_fmaf(t, p, -0.284496736f);
    p = __builtin_fmaf(t, p, 0.254829592f);
    p *= t;
    float e = __expf(-ax * ax);
    float r = __builtin_fmaf(-p, e, 1.0f);   // erf(|x|)
    return __builtin_copysignf(r, x);
}

__global__ __launch_bounds__(NT)
void softrank_ndcg_batch(const float* __restrict__ preds,
                         const float* __restrict__ target,
                         float* __restrict__ ws) {
    extern __shared__ char smem_raw[];
    float*    s_pred = (float*)smem_raw;
    float*    s_red  = (float*)(smem_raw + 4096);
    unsigned* s_cnt  = (unsigned*)(smem_raw + 6144);

    const int b   = blockIdx.x;
    const int tid = threadIdx.x;
    const float* row  = preds  + (size_t)b * LEN;
    const float* trow = target + (size_t)b * LEN;

    if (tid < 8) s_cnt[tid] = 0u;

#if __has_builtin(__builtin_amdgcn_tensor_load_to_lds) && __has_builtin(__builtin_amdgcn_s_wait_tensorcnt)
    // Tensor Data Mover: DMA one 1024x1 f32 tile (4 KB) from global into LDS
    // offset 0. TDM ignores EXEC, so gate by a whole-wave branch (wave 0 only).
    if (tid < 32) {
        unsigned long long ga = (unsigned long long)(uintptr_t)row;
        v4u g0;
        g0[0] = 1u;                                  // count=1 (valid user D#)
        g0[1] = 0u;                                  // lds_addr = 0
        g0[2] = (unsigned)ga;                        // global_addr[31:0]
        g0[3] = ((unsigned)(ga >> 32) & 0x01FFFFFFu) // global_addr[56:32]
                | (2u << 30);                        // type=2 ("image")
        v8i g1;
        g1[0] = (int)(2u << 16);                     // data_size=2 -> 4 bytes
        g1[1] = (int)(1024u << 16);                  // tensor_dim0 = 1024 (lo16)
        g1[2] = (int)(1u << 16);                     // tensor_dim1 = 1
        g1[3] = (int)(1024u << 16);                  // tile_dim0 = 1024
        g1[4] = 1;                                   // tile_dim1 = 1
        g1[5] = 1024;                                // tensor_dim0_stride = 1024
        g1[6] = 0;
        g1[7] = 0;
        v4i g2 = {0, 0, 0, 0};
        v4i g3 = {0, 0, 0, 0};
        v8i g4 = {0, 0, 0, 0, 0, 0, 0, 0};
        __builtin_amdgcn_tensor_load_to_lds(g0, g1, g2, g3, g4, 0);
        __builtin_amdgcn_s_wait_tensorcnt(0);
    }
#else
    for (int k = tid; k < LEN; k += NT) s_pred[k] = row[k];
#endif
    __syncthreads();

    // This thread's 4 list positions.
    const int i0 = tid, i1 = tid + 256, i2 = tid + 512, i3 = tid + 768;
    const float h0 = 0.5f * s_pred[i0];
    const float h1 = 0.5f * s_pred[i1];
    const float h2 = 0.5f * s_pred[i2];
    const float h3 = 0.5f * s_pred[i3];
    const float t0 = trow[i0], t1 = trow[i1], t2 = trow[i2], t3 = trow[i3];

    // Histogram of integer grades 0..4 (the "sort" for ideal DCG).
    atomicAdd(&s_cnt[(int)t0], 1u);
    atomicAdd(&s_cnt[(int)t1], 1u);
    atomicAdd(&s_cnt[(int)t2], 1u);
    atomicAdd(&s_cnt[(int)t3], 1u);

    // Pairwise accumulation: a_k = sum_j erf(0.5*(p_i - p_j)), including j==i
    // (erf(0)=0). ndtr(diff/sqrt2) = 0.5 + 0.5*erf(diff/2), so
    // expected_rank = 1 + (512 + 0.5*a) - 0.5 = 512.5 + 0.5*a.
    float a0 = 0.f, a1 = 0.f, a2 = 0.f, a3 = 0.f;
    const float4* sp4 = (const float4*)s_pred;
    for (int jv = 0; jv < LEN / 4; ++jv) {
        float4 s4 = sp4[jv];
        float sx;
        sx = 0.5f * s4.x;
        a0 += erf_fast(h0 - sx); a1 += erf_fast(h1 - sx);
        a2 += erf_fast(h2 - sx); a3 += erf_fast(h3 - sx);
        sx = 0.5f * s4.y;
        a0 += erf_fast(h0 - sx); a1 += erf_fast(h1 - sx);
        a2 += erf_fast(h2 - sx); a3 += erf_fast(h3 - sx);
        sx = 0.5f * s4.z;
        a0 += erf_fast(h0 - sx); a1 += erf_fast(h1 - sx);
        a2 += erf_fast(h2 - sx); a3 += erf_fast(h3 - sx);
        sx = 0.5f * s4.w;
        a0 += erf_fast(h0 - sx); a1 += erf_fast(h1 - sx);
        a2 += erf_fast(h2 - sx); a3 += erf_fast(h3 - sx);
    }

    // Per-position DCG contributions: gain/log2(er+1), gain = 2^t - 1.
    float dcg_part = 0.f;
    {
        float er, g;
        er = 512.5f + 0.5f * a0; g = (float)((1 << (int)t0) - 1);
        dcg_part += g / __log2f(er + 1.0f);
        er = 512.5f + 0.5f * a1; g = (float)((1 << (int)t1) - 1);
        dcg_part += g / __log2f(er + 1.0f);
        er = 512.5f + 0.5f * a2; g = (float)((1 << (int)t2) - 1);
        dcg_part += g / __log2f(er + 1.0f);
        er = 512.5f + 0.5f * a3; g = (float)((1 << (int)t3) - 1);
        dcg_part += g / __log2f(er + 1.0f);
    }

    __syncthreads();   // histogram complete, safe to read counts

    // Ideal DCG: grade at rank r from cumulative counts (descending grades).
    const int c4 = (int)s_cnt[4];
    const int b3 = c4 + (int)s_cnt[3];
    const int b2 = b3 + (int)s_cnt[2];
    const int b1 = b2 + (int)s_cnt[1];
    float idcg_part = 0.f;
    #pragma unroll
    for (int k = 0; k < 4; ++k) {
        int r = tid + NT * k;
        int g = (r < c4) ? 4 : (r < b3) ? 3 : (r < b2) ? 2 : (r < b1) ? 1 : 0;
        idcg_part += (float)((1 << g) - 1) / __log2f((float)(r + 2));
    }

    // Deterministic block tree-reduction of dcg and idcg simultaneously.
    s_red[tid]      = dcg_part;
    s_red[NT + tid] = idcg_part;
    __syncthreads();
    #pragma unroll
    for (int s = NT / 2; s > 0; s >>= 1) {
        if (tid < s) {
            s_red[tid]      += s_red[tid + s];
            s_red[NT + tid] += s_red[NT + tid + s];
        }
        __syncthreads();
    }
    if (tid == 0) ws[b] = s_red[0] / (s_red[NT] + 1e-10f);
}

__global__ __launch_bounds__(BATCH)
void softrank_ndcg_finalize(const float* __restrict__ ws, float* __restrict__ out) {
    __shared__ float red[BATCH];
    const int tid = threadIdx.x;
    red[tid] = ws[tid];
    __syncthreads();
    #pragma unroll
    for (int s = BATCH / 2; s > 0; s >>= 1) {
        if (tid < s) red[tid] += red[tid + s];
        __syncthreads();
    }
    if (tid == 0) out[0] = -red[0] / (float)BATCH;
}

extern "C" void kernel_launch(void* const* d_in, const int* in_sizes, int n_in,
                              void* d_out, int out_size, void* d_ws, size_t ws_size,
                              hipStream_t stream) {
    const float* preds  = (const float*)d_in[0];
    const float* target = (const float*)d_in[1];
    float* ws  = (float*)d_ws;     // 128 per-batch ndcg values
    float* out = (float*)d_out;    // scalar

    softrank_ndcg_batch<<<BATCH, NT, SMEM_BYTES, stream>>>(preds, target, ws);
    softrank_ndcg_finalize<<<1, BATCH, 0, stream>>>(ws, out);
}